// ExpandedStandardFMNet_36395552866818
// MI455X (gfx1250) — compile-verified
//
#include <hip/hip_runtime.h>

// ---------------------------------------------------------------------------
// ExpandedStandardFMNet functional-map solve, restructured via Kronecker
// algebra for MI455X (gfx1250, wave32, WMMA f32 16x16x4).
//
//   A   = tx @ fx                      [64 x 256]   (WMMA, K=5000)
//   TyF = ty @ fy                      [64 x 256]   (WMMA, K=5000)
//   B   = sy @ TyF                     [64 x 256]
//   isx = inv(sx)                      (Gauss-Jordan, 1 workgroup)
//   G   = sy^T sy
//   E1  = A A^T + l * isx^T Lx^2 isx
//   E2  = -l * isx^T Lx isx            (symmetric => E3 == E2)
//   E4  =  l * isx^T isx
//   R   = A B^T sy                     (rhs reshaped [64 x 64])
//   solve (sum of 4 Kron terms) * vec(X) = vec(R) via CG; op-matvec =
//     E1*(X G) + E2*((X.ey) G) + E2*((X G).ey) + E4*(((X.ey) G).ey)
//   out = X^T
// ---------------------------------------------------------------------------

typedef __attribute__((ext_vector_type(2))) float v2f;
typedef __attribute__((ext_vector_type(8))) float v8f;

#define KD 64
#define MD 256
#define VD 5000
#define LMB 0.001f
#define CG_ITERS 256

__device__ __forceinline__ v8f wmma_f32(v2f a, v2f b, v8f c) {
  // (neg_a, A, neg_b, B, c_mod, C, reuse_a, reuse_b)
  return __builtin_amdgcn_wmma_f32_16x16x4_f32(false, a, false, b, (short)0, c,
                                               false, false);
}

// --------------------------- A = X @ Y (64 x VD x 256), batched z ----------
__global__ void __launch_bounds__(32)
spectral_gemm(const float* __restrict__ X0, const float* __restrict__ Y0,
              float* __restrict__ C0, const float* __restrict__ X1,
              const float* __restrict__ Y1, float* __restrict__ C1) {
  const float* X = blockIdx.z ? X1 : X0;
  const float* Y = blockIdx.z ? Y1 : Y0;
  float* C = blockIdx.z ? C1 : C0;
  const int tn = blockIdx.x;       // 0..15  (N tiles of 16)
  const int tm = blockIdx.y;       // 0..3   (M tiles of 16)
  const int lane = threadIdx.x;    // one wave32 per 16x16 tile
  const int mr = lane & 15;
  const int ks = (lane >> 4) << 1; // K pair: lanes 0-15 -> {0,1}, 16-31 -> {2,3}
  const float* xrow = X + (size_t)(tm * 16 + mr) * VD;
  v8f acc = {};
  for (int kk = 0; kk < VD; kk += 4) {
    v2f a, b;
    a.x = xrow[kk + ks];
    a.y = xrow[kk + ks + 1];
    const int nb = tn * 16 + mr;
    b.x = Y[(size_t)(kk + ks) * MD + nb];
    b.y = Y[(size_t)(kk + ks + 1) * MD + nb];
    acc = wmma_f32(a, b, acc);
  }
  const int rb = (lane >> 4) * 8;  // C layout: VGPR v -> row v (+8 for hi half)
#pragma unroll
  for (int v = 0; v < 8; ++v)
    C[(size_t)(tm * 16 + rb + v) * MD + tn * 16 + mr] = acc[v];
}

// --------------------------- 64x64 Gauss-Jordan inverse --------------------
__global__ void __launch_bounds__(256)
invert64(const float* __restrict__ S, float* __restrict__ Sinv) {
  __shared__ float aug[64][128];
  __shared__ float fac[64];
  const int tid = threadIdx.x;
  for (int e = tid; e < 64 * 128; e += 256) {
    const int r = e >> 7, c = e & 127;
    aug[r][c] = (c < 64) ? S[r * 64 + c] : ((c - 64 == r) ? 1.f : 0.f);
  }
  __syncthreads();
  for (int p = 0; p < 64; ++p) {
    const float pinv = 1.0f / aug[p][p];
    __syncthreads();
    for (int c = tid; c < 128; c += 256) aug[p][c] *= pinv;
    __syncthreads();
    if (tid < 64) fac[tid] = aug[tid][p];
    __syncthreads();
    for (int e = tid; e < 64 * 128; e += 256) {
      const int r = e >> 7, c = e & 127;
      if (r != p) aug[r][c] -= fac[r] * aug[p][c];
    }
    __syncthreads();
  }
  for (int e = tid; e < 64 * 64; e += 256)
    Sinv[e] = aug[e >> 6][64 + (e & 63)];
}

// --------------------------- B = sy @ TyF  [64 x 256] ----------------------
__global__ void __launch_bounds__(256)
glue_B(const float* __restrict__ sy, const float* __restrict__ TyF,
       float* __restrict__ B) {
  const int idx = blockIdx.x * 256 + threadIdx.x;  // 16384
  const int i = idx >> 8, j = idx & 255;
  float s = 0.f;
  for (int k = 0; k < 64; ++k) s += sy[i * 64 + k] * TyF[k * 256 + j];
  B[idx] = s;
}

// ----------------- G, P1 = A B^T, E1/E2/E4 (all 64x64) ---------------------
__global__ void __launch_bounds__(256)
glue_small(const float* __restrict__ sy, const float* __restrict__ A,
           const float* __restrict__ B, const float* __restrict__ ex,
           const float* __restrict__ isx, float* __restrict__ G,
           float* __restrict__ P1, float* __restrict__ E1,
           float* __restrict__ E2, float* __restrict__ E4) {
  const int idx = blockIdx.x * 256 + threadIdx.x;  // 4096
  const int i = idx >> 6, j = idx & 63;
  float g = 0.f, s1 = 0.f, s2 = 0.f, s4 = 0.f;
  for (int k = 0; k < 64; ++k) {
    g += sy[k * 64 + i] * sy[k * 64 + j];
    const float u = isx[k * 64 + i] * isx[k * 64 + j];
    const float e = ex[k];
    s1 += e * e * u;
    s2 += e * u;
    s4 += u;
  }
  float aat = 0.f, p1 = 0.f;
  for (int m = 0; m < 256; ++m) {
    const float am = A[i * 256 + m];
    aat += am * A[j * 256 + m];
    p1 += am * B[j * 256 + m];
  }
  G[idx] = g;
  P1[idx] = p1;
  E1[idx] = aat + LMB * s1;
  E2[idx] = -LMB * s2;  // symmetric: also serves as E3
  E4[idx] = LMB * s4;
}

// --------------------------- R = P1 @ sy -----------------------------------
__global__ void __launch_bounds__(256)
glue_rhs(const float* __restrict__ P1, const float* __restrict__ sy,
         float* __restrict__ R) {
  const int idx = blockIdx.x * 256 + threadIdx.x;
  const int i = idx >> 6, j = idx & 63;
  float s = 0.f;
  for (int k = 0; k < 64; ++k) s += P1[i * 64 + k] * sy[k * 64 + j];
  R[idx] = s;
}

// --------------------------- CG solve (one WGP, LDS-resident) --------------
__device__ __forceinline__ float block_sum(float v, float* red) {
  const int tid = threadIdx.x;
  red[tid] = v;
  __syncthreads();
#pragma unroll
  for (int s = 128; s > 0; s >>= 1) {
    if (tid < s) red[tid] += red[tid + s];
    __syncthreads();
  }
  const float r = red[0];
  __syncthreads();
  return r;
}

__global__ void __launch_bounds__(256)
cg_solve(const float* __restrict__ E1, const float* __restrict__ E2,
         const float* __restrict__ E4, const float* __restrict__ G,
         const float* __restrict__ ey, const float* __restrict__ R,
         float* __restrict__ out) {
  extern __shared__ float sm[];
  float* sE = sm;             // 4 * 4096  (E1, E2, E3==E2, E4)
  float* sG = sE + 16384;     // 4096
  float* sP = sG + 4096;      // 4096  CG direction, reshaped 64x64
  float* sW1 = sP + 4096;     // 4096  W1 = P G
  float* sW2 = sW1 + 4096;    // 4096  W2 = (P .* ey_col) G
  float* sQ = sW2 + 4096;     // 4096  q = op * p
  float* sEy = sQ + 4096;     // 64
  float* sRed = sEy + 64;     // 256
  const int tid = threadIdx.x;

  for (int i = tid; i < 4096; i += 256) {
    sE[i] = E1[i];
    const float e2 = E2[i];
    sE[4096 + i] = e2;
    sE[8192 + i] = e2;        // E3 == E2 (isx^T Lx isx symmetric)
    sE[12288 + i] = E4[i];
    sG[i] = G[i];
  }
  if (tid < 64) sEy[tid] = ey[tid];

  const int row = tid >> 2;           // each thread owns 16 contiguous elems
  const int cb = (tid & 3) << 4;
  float xloc[16], rloc[16];
#pragma unroll
  for (int e = 0; e < 16; ++e) {
    const float rv = R[row * 64 + cb + e];
    xloc[e] = 0.f;
    rloc[e] = rv;
    sP[row * 64 + cb + e] = rv;
  }
  __syncthreads();

  float part = 0.f;
#pragma unroll
  for (int e = 0; e < 16; ++e) part += rloc[e] * rloc[e];
  float rs_old = block_sum(part, sRed);

  const int lane = tid & 31, wave = tid >> 5;
  const int mr = lane & 15;
  const int ksl = (lane >> 4) << 1;
  const int rb = (lane >> 4) * 8;

  for (int it = 0; it < CG_ITERS; ++it) {
    // ---- W1 = P G, W2 = (P .* ey_col) G : 8 waves x 2 tiles each ----
#pragma unroll
    for (int tt = 0; tt < 2; ++tt) {
      const int t = wave * 2 + tt;
      const int tm = t >> 2, tn = t & 3;
      v8f a1 = {}, a2 = {};
      for (int kk = 0; kk < 64; kk += 4) {
        v2f a, ae, b;
        const int r0 = (tm * 16 + mr) * 64 + kk + ksl;
        a.x = sP[r0];
        a.y = sP[r0 + 1];
        ae.x = a.x * sEy[kk + ksl];
        ae.y = a.y * sEy[kk + ksl + 1];
        const int b0 = (kk + ksl) * 64 + tn * 16 + mr;
        b.x = sG[b0];
        b.y = sG[b0 + 64];
        a1 = wmma_f32(a, b, a1);
        a2 = wmma_f32(ae, b, a2);
      }
#pragma unroll
      for (int v = 0; v < 8; ++v) {
        const int o = (tm * 16 + rb + v) * 64 + tn * 16 + mr;
        sW1[o] = a1[v];
        sW2[o] = a2[v];
      }
    }
    __syncthreads();

    // ---- Q = E1 W1 + E2 W2 + E3 (W1.*ey) + E4 (W2.*ey) ----
#pragma unroll
    for (int tt = 0; tt < 2; ++tt) {
      const int t = wave * 2 + tt;
      const int tm = t >> 2, tn = t & 3;
      const float eyn = sEy[tn * 16 + mr];  // column scale, lane-constant
      v8f acc = {};
      for (int kk = 0; kk < 64; kk += 4) {
        const int ra = (tm * 16 + mr) * 64 + kk + ksl;
        v2f a1v, a2v, a3v, a4v, b1, b2, b3, b4;
        a1v.x = sE[ra];          a1v.y = sE[ra + 1];
        a2v.x = sE[4096 + ra];   a2v.y = sE[4096 + ra + 1];
        a3v.x = sE[8192 + ra];   a3v.y = sE[8192 + ra + 1];
        a4v.x = sE[12288 + ra];  a4v.y = sE[12288 + ra + 1];
        const int b0 = (kk + ksl) * 64 + tn * 16 + mr;
        b1.x = sW1[b0];  b1.y = sW1[b0 + 64];
        b2.x = sW2[b0];  b2.y = sW2[b0 + 64];
        b3.x = b1.x * eyn;  b3.y = b1.y * eyn;
        b4.x = b2.x * eyn;  b4.y = b2.y * eyn;
        acc = wmma_f32(a1v, b1, acc);
        acc = wmma_f32(a2v, b2, acc);
        acc = wmma_f32(a3v, b3, acc);
        acc = wmma_f32(a4v, b4, acc);
      }
#pragma unroll
      for (int v = 0; v < 8; ++v)
        sQ[(tm * 16 + rb + v) * 64 + tn * 16 + mr] = acc[v];
    }
    __syncthreads();

    // ---- scalar CG updates ----
    part = 0.f;
#pragma unroll
    for (int e = 0; e < 16; ++e) {
      const int o = row * 64 + cb + e;
      part += sP[o] * sQ[o];
    }
    const float pq = block_sum(part, sRed);
    const float alpha = rs_old / pq;
    part = 0.f;
#pragma unroll
    for (int e = 0; e < 16; ++e) {
      const int o = row * 64 + cb + e;
      xloc[e] += alpha * sP[o];
      rloc[e] -= alpha * sQ[o];
      part += rloc[e] * rloc[e];
    }
    const float rs_new = block_sum(part, sRed);
    const float beta = rs_new / rs_old;
    rs_old = rs_new;
#pragma unroll
    for (int e = 0; e < 16; ++e) {
      const int o = row * 64 + cb + e;
      sP[o] = rloc[e] + beta * sP[o];
    }
    __syncthreads();
  }

  // out = X^T : out[j2*64 + j1] = X[j1][j2]
#pragma unroll
  for (int e = 0; e < 16; ++e) out[(cb + e) * 64 + row] = xloc[e];
}

// ---------------------------------------------------------------------------
extern "C" void kernel_launch(void* const* d_in, const int* in_sizes, int n_in,
                              void* d_out, int out_size, void* d_ws,
                              size_t ws_size, hipStream_t stream) {
  (void)in_sizes; (void)n_in; (void)out_size; (void)ws_size;
  const float* fx = (const float*)d_in[0];  // [5000,256]
  const float* fy = (const float*)d_in[1];
  const float* ex = (const float*)d_in[2];  // [64]
  const float* ey = (const float*)d_in[3];
  const float* tx = (const float*)d_in[4];  // [64,5000]
  const float* ty = (const float*)d_in[5];
  const float* sx = (const float*)d_in[6];  // [64,64]
  const float* sy = (const float*)d_in[7];

  float* ws = (float*)d_ws;
  float* A   = ws;             // 16384
  float* TyF = ws + 16384;     // 16384
  float* Bm  = ws + 32768;     // 16384
  float* isx = ws + 49152;     // 4096
  float* G   = ws + 53248;     // 4096
  float* P1  = ws + 57344;     // 4096
  float* E1  = ws + 61440;     // 4096
  float* E2  = ws + 65536;     // 4096
  float* E4  = ws + 69632;     // 4096
  float* R   = ws + 73728;     // 4096  (end: 77824 floats = 311 KB)

  spectral_gemm<<<dim3(16, 4, 2), 32, 0, stream>>>(tx, fx, A, ty, fy, TyF);
  invert64<<<1, 256, 0, stream>>>(sx, isx);
  glue_B<<<64, 256, 0, stream>>>(sy, TyF, Bm);
  glue_small<<<16, 256, 0, stream>>>(sy, A, Bm, ex, isx, G, P1, E1, E2, E4);
  glue_rhs<<<16, 256, 0, stream>>>(P1, sy, R);
  const size_t smem = (size_t)(16384 + 5 * 4096 + 64 + 256) * sizeof(float);
  cg_solve<<<1, 256, smem, stream>>>(E1, E2, E4, G, ey, R, (float*)d_out);
}